// ExampleNet_6545530159791
// MI455X (gfx1250) — compile-verified
//
#include <hip/hip_runtime.h>

typedef float v2f __attribute__((ext_vector_type(2)));
typedef float v8f __attribute__((ext_vector_type(8)));

#define NPTS  80000
#define CIN   3
#define COUT  32
#define KVOL  125
#define WSZ   (KVOL * CIN * COUT)   // 12000 floats = 48 KB
#define MPAD  950000
#define NTILES (MPAD / 16)          // 59375, exact

__global__ void spconv_zero_kernel(float* __restrict__ out, int n) {
  int i = blockIdx.x * blockDim.x + threadIdx.x;
  int stride = gridDim.x * blockDim.x;
  for (; i < n; i += stride) out[i] = 0.0f;
}

__global__ __launch_bounds__(256) void spconv_wmma_kernel(
    const float* __restrict__ feat,     // [NPTS, 3]
    const float* __restrict__ weight,   // [125, 3, 32]
    const int*   __restrict__ in_idx,   // [MPAD]
    const int*   __restrict__ out_idx,  // [MPAD]
    const int*   __restrict__ off_idx,  // [MPAD], non-decreasing
    float*       __restrict__ out) {    // [NPTS, 32]
  __shared__ float sW[WSZ];
  // Stage full weight tensor in LDS (48 KB of 320 KB WGP LDS); L2-resident source.
  for (int i = threadIdx.x; i < WSZ; i += blockDim.x) sW[i] = weight[i];
  __syncthreads();

  const int lane = threadIdx.x & 31;   // wave32
  const int row  = lane & 15;          // tile row (M) this lane serves
  const int hi   = lane >> 4;          // 0: K=0/1 half, 1: K=2/3 half (A layout)
  const int waveId = (blockIdx.x * blockDim.x + threadIdx.x) >> 5;
  const int nWaves = (gridDim.x * blockDim.x) >> 5;

  for (int tile = waveId; tile < NTILES; tile += nWaves) {
    const int base = tile << 4;

    // Prefetch the indices of this wave's next tile (global_prefetch_b8).
    const int nbase = base + (nWaves << 4);
    if (nbase < MPAD) {
      __builtin_prefetch(in_idx  + nbase, 0, 0);
      __builtin_prefetch(out_idx + nbase, 0, 0);
    }

    // off_idx is sorted: first==last  =>  all 16 entries share one weight slice.
    const int off0  = off_idx[base];
    const int off15 = off_idx[base + 15];

    if (off0 == off15) {
      // ---- WMMA path: D(16x16) = A(16x4) x B(4x16) + 0, fp32, two halves of COUT ----
      // A layout (ISA 7.12.2, 32-bit A 16x4): lanes 0-15 -> VGPR0=K0, VGPR1=K1;
      // lanes 16-31 -> VGPR0=K2, VGPR1=K3 (K=3 column is the CIN pad -> 0).
      const int gi = in_idx[base + row];
      v2f a;
      a.x = feat[gi * CIN + (hi ? 2 : 0)];
      a.y = hi ? 0.0f : feat[gi * CIN + 1];

      const float* wb = &sW[off0 * (CIN * COUT)];
      v8f c = {};
#pragma unroll
      for (int h = 0; h < 2; ++h) {
        const int ch = (h << 4) + row;  // output channel this lane's B column covers
        // B layout mirrors A along K: VGPR0 = {K0 | K2}, VGPR1 = {K1 | K3(pad)}.
        v2f b;
        b.x = wb[(hi ? 2 : 0) * COUT + ch];
        b.y = hi ? 0.0f : wb[1 * COUT + ch];

        v8f d = __builtin_amdgcn_wmma_f32_16x16x4_f32(
            /*neg_a=*/false, a, /*neg_b=*/false, b,
            /*c_mod=*/(short)0, c, /*reuse_a=*/false, /*reuse_b=*/false);

        // D layout: VGPR r holds row r (lanes 0-15) / row r+8 (lanes 16-31), col = lane&15.
#pragma unroll
        for (int r = 0; r < 8; ++r) {
          const int trow = r + (hi << 3);
          const int orow = out_idx[base + trow];
          if (orow < NPTS)
            unsafeAtomicAdd(&out[orow * COUT + ch], d[r]);  // global_atomic_add_f32
        }
      }
    } else {
      // ---- Boundary tile (<=125 of 59375): scalar fallback, full precision ----
      // lane handles entry base+row, channels [hi*16, hi*16+16).
      const int m  = base + row;
      const int gi = in_idx[m];
      const int oo = out_idx[m];
      const int of = off_idx[m];
      if (oo < NPTS) {
        const float f0 = feat[gi * CIN + 0];
        const float f1 = feat[gi * CIN + 1];
        const float f2 = feat[gi * CIN + 2];
        const float* wb = &sW[of * (CIN * COUT)];
#pragma unroll
        for (int cc = 0; cc < 16; ++cc) {
          const int ch = (hi << 4) + cc;
          const float v = f0 * wb[ch] + f1 * wb[COUT + ch] + f2 * wb[2 * COUT + ch];
          unsafeAtomicAdd(&out[oo * COUT + ch], v);
        }
      }
    }
  }
}

extern "C" void kernel_launch(void* const* d_in, const int* in_sizes, int n_in,
                              void* d_out, int out_size, void* d_ws, size_t ws_size,
                              hipStream_t stream) {
  const float* feat    = (const float*)d_in[0];
  const float* weight  = (const float*)d_in[1];
  const int*   in_idx  = (const int*)d_in[2];
  const int*   out_idx = (const int*)d_in[3];
  const int*   off_idx = (const int*)d_in[4];
  float* out = (float*)d_out;

  // Zero the accumulator output (harness poisons it; we scatter-add into it).
  spconv_zero_kernel<<<512, 256, 0, stream>>>(out, out_size);

  // 1024 blocks x 8 wave32s = 8192 waves grid-striding over 59375 tiles.
  spconv_wmma_kernel<<<1024, 256, 0, stream>>>(feat, weight, in_idx, out_idx,
                                               off_idx, out);
}